// GConv_78709570667298
// MI455X (gfx1250) — compile-verified
//
#include <hip/hip_runtime.h>

typedef float v2f __attribute__((ext_vector_type(2)));
typedef float v8f __attribute__((ext_vector_type(8)));

#define N_NODES 10000
#define DIN     256
#define DOUT    256
#define BATCH   2
#define M_TOTAL (BATCH * N_NODES)   // 20000 rows
#define ROWS_PER_BLOCK 160          // 10 wave32s x 16-row tiles
#define THREADS 320                 // 10 waves
#define SLABS   (M_TOTAL / ROWS_PER_BLOCK)   // 125, exact
#define NTILES  (DOUT / 16)                  // 16 column tiles

// ---------------------------------------------------------------------------
// Kernel 1: fused dual GEMM  x @ W  and  x @ loop_W  via V_WMMA_F32_16X16X4_F32.
// Block stages the 256x16 K-panels of BOTH weight matrices in LDS (column-
// major, 32 KB) so each lane's B fragment is one ds_load_b64. Each wave owns a
// 16-row tile and produces both outputs from a single A fragment:
//   per k-step: 1 global b64 (A) + 2 ds b64 (B,B') + 2 WMMA.
// support -> workspace; support_loop + bias -> d_out (re-inits d_out).
// ---------------------------------------------------------------------------
__global__ __launch_bounds__(THREADS) void gconv_gemm_wmma(
    const float* __restrict__ x, const float* __restrict__ W,
    const float* __restrict__ loopW, const float* __restrict__ bias,
    float* __restrict__ support, float* __restrict__ out)
{
    __shared__ float ldsW[DIN * 16];   // [col][k] column-major panel of W
    __shared__ float ldsL[DIN * 16];   // [col][k] column-major panel of loop_W

    const int tid  = threadIdx.x;
    const int wave = tid >> 5;          // 0..9
    const int lane = tid & 31;
    const int hi   = lane >> 4;         // K offset selector (+2 for lanes 16-31)
    const int lo   = lane & 15;         // M (A) / N (B,C,D) index

    const int n0   = blockIdx.x * 16;                  // column tile base
    const int row0 = blockIdx.y * ROWS_PER_BLOCK + wave * 16;

    // Cooperative panel load: i = k*16 + col -> coalesced 64B global segments,
    // stored column-major so K is contiguous in LDS.
    for (int i = tid; i < DIN * 16; i += THREADS) {
        const int k   = i >> 4;
        const int col = i & 15;
        ldsW[col * DIN + k] = W    [(size_t)k * DOUT + n0 + col];
        ldsL[col * DIN + k] = loopW[(size_t)k * DOUT + n0 + col];
    }
    __syncthreads();

    // A: 16x4 f32 tile. lane(0-15): row=lo, {K,K+1}; lane(16-31): {K+2,K+3}
    const float* Arow = x + (size_t)(row0 + lo) * DIN;
    const float* bWp  = &ldsW[lo * DIN];
    const float* bLp  = &ldsL[lo * DIN];

    v8f accW = {};
    v8f accL = {};
    #pragma unroll 8
    for (int k = 0; k < DIN; k += 4) {
        const int kk = k + 2 * hi;
        v2f a  = *(const v2f*)(Arow + kk);   // global b64, 8B aligned
        v2f bw = *(const v2f*)(bWp + kk);    // ds b64, 8B aligned
        v2f bl = *(const v2f*)(bLp + kk);    // ds b64, 8B aligned
        // 8 args: (neg_a, A, neg_b, B, c_mod, C, reuse_a, reuse_b)
        accW = __builtin_amdgcn_wmma_f32_16x16x4_f32(
            false, a, false, bw, (short)0, accW, false, false);
        accL = __builtin_amdgcn_wmma_f32_16x16x4_f32(
            false, a, false, bl, (short)0, accL, false, false);
    }

    // D layout: VGPR r holds row (r + 8*hi), column (n0 + lo)
    const float bv = bias[n0 + lo];
    float* dstS = support + (size_t)(row0 + 8 * hi) * DOUT + n0 + lo;
    float* dstO = out     + (size_t)(row0 + 8 * hi) * DOUT + n0 + lo;
    #pragma unroll
    for (int r = 0; r < 8; ++r) {
        dstS[(size_t)r * DOUT] = accW[r];
        dstO[(size_t)r * DOUT] = accL[r] + bv;
    }
}

// ---------------------------------------------------------------------------
// Kernel 2: COO SpMM scatter. One block per edge, thread d = feature index.
// out[b, rows[e], d] += edge_vals[e] * support[b, cols[e], d]
// support (20.5 MB) is L2-resident; atomics resolve at L2.
// ---------------------------------------------------------------------------
__global__ __launch_bounds__(256) void gconv_spmm_edges(
    const float* __restrict__ support, const float* __restrict__ edge_vals,
    const int* __restrict__ rows, const int* __restrict__ cols,
    float* __restrict__ out, int E)
{
    const int e = blockIdx.x;
    if (e >= E) return;
    const int d = threadIdx.x;               // 0..255
    const float v = edge_vals[e];
    const int r = rows[e];
    const int c = cols[e];
    #pragma unroll
    for (int b = 0; b < BATCH; ++b) {
        const float m = v * support[((size_t)b * N_NODES + c) * DOUT + d];
        unsafeAtomicAdd(&out[((size_t)b * N_NODES + r) * DOUT + d], m);
    }
}

// ---------------------------------------------------------------------------
// Kernel 3: in-place ReLU after aggregation.
// ---------------------------------------------------------------------------
__global__ __launch_bounds__(256) void gconv_relu(float* __restrict__ out, int n)
{
    const int i = blockIdx.x * blockDim.x + threadIdx.x;
    if (i < n) {
        const float v = out[i];
        out[i] = v > 0.0f ? v : 0.0f;
    }
}

extern "C" void kernel_launch(void* const* d_in, const int* in_sizes, int n_in,
                              void* d_out, int out_size, void* d_ws, size_t ws_size,
                              hipStream_t stream)
{
    const float* x         = (const float*)d_in[0];
    const float* W         = (const float*)d_in[1];
    const float* loopW     = (const float*)d_in[2];
    const float* bias      = (const float*)d_in[3];
    const float* edge_vals = (const float*)d_in[4];
    const int*   rows      = (const int*)d_in[5];
    const int*   cols      = (const int*)d_in[6];
    float* out     = (float*)d_out;
    float* support = (float*)d_ws;           // 2*10000*256*4 = 20.48 MB
    const int E = in_sizes[4];

    // Dual GEMM: grid = (16 column tiles) x (125 row slabs), 320 threads/block
    dim3 grid(NTILES, SLABS);
    gconv_gemm_wmma<<<grid, THREADS, 0, stream>>>(x, W, loopW, bias, support, out);

    // SpMM scatter over edges
    gconv_spmm_edges<<<E, 256, 0, stream>>>(support, edge_vals, rows, cols, out, E);

    // ReLU
    const int n = BATCH * N_NODES * DOUT;
    gconv_relu<<<(n + 255) / 256, 256, 0, stream>>>(out, n);
}